// DTCAModel_39195871543543
// MI455X (gfx1250) — compile-verified
//
#include <hip/hip_runtime.h>
#include <hip/hip_bf16.h>
#include <math.h>

typedef __attribute__((ext_vector_type(2))) float v2f;
typedef __attribute__((ext_vector_type(8))) float v8f;

#define BATCH 64
#define MDIM  512
#define NDIM  512
#define DDIM  768
#define TSTEP 512
#define KTAG  21
#define OT_ITERS 50

static const size_t SZ_MAT = (size_t)BATCH * MDIM * NDIM;

// ---- CDNA5 async global->LDS (ASYNCcnt-tracked) ------------------------
__device__ __forceinline__ void async_b128(const float* g, const float* lds) {
  unsigned int loff = (unsigned int)(unsigned long long)(uintptr_t)lds;
  asm volatile("global_load_async_to_lds_b128 %0, %1, off"
               :: "v"(loff), "v"((unsigned long long)(uintptr_t)g)
               : "memory");
}
__device__ __forceinline__ void wait_async_le8() {
  asm volatile("s_wait_asynccnt 0x8" ::: "memory");
}
__device__ __forceinline__ void wait_async_0() {
  asm volatile("s_wait_asynccnt 0x0" ::: "memory");
}

// =========================================================================
// 1) inverse L2 norms for txt (rows 0..B*M-1) and img (rows B*M..)
// =========================================================================
__global__ void invnorm_kernel(const float* __restrict__ x,
                               const float* __restrict__ y,
                               float* __restrict__ inv_x,
                               float* __restrict__ inv_y) {
  int wave = (blockIdx.x * (blockDim.x >> 5)) + (threadIdx.x >> 5);
  int lane = threadIdx.x & 31;
  const int half_rows = BATCH * MDIM;
  const float* src;
  float* dst;
  int row;
  if (wave < half_rows) { row = wave;              src = x; dst = inv_x; }
  else                  { row = wave - half_rows;  src = y; dst = inv_y; }
  const float* p = src + (size_t)row * DDIM;
  float acc = 0.f;
  for (int k = lane; k < DDIM; k += 32) { float v = p[k]; acc += v * v; }
  for (int off = 16; off > 0; off >>= 1) acc += __shfl_down(acc, off, 32);
  if (lane == 0) dst[row] = 1.0f / fmaxf(sqrtf(acc), 1e-5f);
}

// =========================================================================
// 2) seq lengths (float) + sigma init: sigma[b][m] = pad ? 0 : 1/xlen
// =========================================================================
__global__ void lengths_sigma_kernel(const unsigned char* __restrict__ txt_pad,
                                     const unsigned char* __restrict__ img_pad,
                                     float* __restrict__ xlen,
                                     float* __restrict__ ylen,
                                     float* __restrict__ sigma) {
  __shared__ int sx[512];
  __shared__ int sy[512];
  int b = blockIdx.x, t = threadIdx.x;
  sx[t] = txt_pad[b * MDIM + t] ? 1 : 0;
  sy[t] = img_pad[b * NDIM + t] ? 1 : 0;
  __syncthreads();
  for (int off = 256; off > 0; off >>= 1) {
    if (t < off) { sx[t] += sx[t + off]; sy[t] += sy[t + off]; }
    __syncthreads();
  }
  float xl = (float)(MDIM - sx[0]);
  if (t == 0) { xlen[b] = xl; ylen[b] = (float)(NDIM - sy[0]); }
  sigma[b * MDIM + t] = txt_pad[b * MDIM + t] ? 0.f : (1.0f / xl);
}

// =========================================================================
// 3) batched cost GEMM via V_WMMA_F32_16X16X4_F32
//    block = 128 thr (4 waves, 2x2 wave grid), 64x64 block tile,
//    32x32 register tile per wave (2x2 accumulators -> 2x A/B reuse),
//    K panels of 32 double-buffered via GLOBAL_LOAD_ASYNC_TO_LDS_B128.
//    Epilogue: cost (masked), A = exp(-cost/beta) masked, T init.
// =========================================================================
#define KPAD 36   // float stride per LDS row: 144B, 16B-aligned, conflict-free

__global__ void __launch_bounds__(128)
cost_wmma_kernel(const float* __restrict__ x, const float* __restrict__ y,
                 const float* __restrict__ inv_x, const float* __restrict__ inv_y,
                 const unsigned char* __restrict__ txt_pad,
                 const unsigned char* __restrict__ img_pad,
                 float* __restrict__ cost, float* __restrict__ Amat,
                 float* __restrict__ Tmat) {
  __shared__ float Xs[2][64][KPAD];
  __shared__ float Ys[2][64][KPAD];
  const int b  = blockIdx.z;
  const int bm = blockIdx.x * 64;
  const int bn = blockIdx.y * 64;
  const int tid  = threadIdx.x;
  const int wave = tid >> 5;
  const int lane = tid & 31;
  const int wmy = (wave >> 1) * 32;      // wave tile m offset in block tile
  const int wnx = (wave & 1) * 32;       // wave tile n offset
  const int half = lane >> 4;            // selects K pair {0,1} vs {2,3}
  const int lr   = lane & 15;

  // async loader mapping: 64 rows x 32 cols per matrix, 128 threads,
  // 16 floats (4 x b128) per thread per matrix
  const int ldr = tid >> 1;              // 0..63
  const int ldc = (tid & 1) * 16;        // 0 or 16

  const float* xrow = x + ((size_t)b * MDIM + bm + ldr) * DDIM + ldc;
  const float* yrow = y + ((size_t)b * NDIM + bn + ldr) * DDIM + ldc;

  v8f acc00 = {}, acc01 = {}, acc10 = {}, acc11 = {};

  // prologue: panel 0 into buffer 0
#pragma unroll
  for (int i = 0; i < 4; ++i) {
    async_b128(xrow + i * 4, &Xs[0][ldr][ldc + i * 4]);
    async_b128(yrow + i * 4, &Ys[0][ldr][ldc + i * 4]);
  }

  const int NPANEL = DDIM / 32;          // 24
  for (int p = 0; p < NPANEL; ++p) {
    const int cur = p & 1;
    if (p + 1 < NPANEL) {
      const int nxt = (p + 1) & 1;
      const int kb = (p + 1) * 32;
#pragma unroll
      for (int i = 0; i < 4; ++i) {
        async_b128(xrow + kb + i * 4, &Xs[nxt][ldr][ldc + i * 4]);
        async_b128(yrow + kb + i * 4, &Ys[nxt][ldr][ldc + i * 4]);
      }
      wait_async_le8();                  // in-order: panel p landed
    } else {
      wait_async_0();
    }
    __syncthreads();                     // panel p visible to all waves
#pragma unroll
    for (int kk = 0; kk < 32; kk += 4) {
      const int kc = kk + 2 * half;
      v2f a0, a1, b0, b1;
      a0.x = Xs[cur][wmy + lr][kc];      a0.y = Xs[cur][wmy + lr][kc + 1];
      a1.x = Xs[cur][wmy + 16 + lr][kc]; a1.y = Xs[cur][wmy + 16 + lr][kc + 1];
      b0.x = Ys[cur][wnx + lr][kc];      b0.y = Ys[cur][wnx + lr][kc + 1];
      b1.x = Ys[cur][wnx + 16 + lr][kc]; b1.y = Ys[cur][wnx + 16 + lr][kc + 1];
      acc00 = __builtin_amdgcn_wmma_f32_16x16x4_f32(false, a0, false, b0, (short)0, acc00, false, false);
      acc01 = __builtin_amdgcn_wmma_f32_16x16x4_f32(false, a0, false, b1, (short)0, acc01, false, false);
      acc10 = __builtin_amdgcn_wmma_f32_16x16x4_f32(false, a1, false, b0, (short)0, acc10, false, false);
      acc11 = __builtin_amdgcn_wmma_f32_16x16x4_f32(false, a1, false, b1, (short)0, acc11, false, false);
    }
    __syncthreads();                     // all reads of buffer done before reuse
  }

  // epilogue: D layout — VGPR r: M = r + 8*half, N = lane%16
#pragma unroll
  for (int sm = 0; sm < 2; ++sm) {
#pragma unroll
    for (int sn = 0; sn < 2; ++sn) {
      const v8f acc = (sm == 0) ? (sn == 0 ? acc00 : acc01)
                                : (sn == 0 ? acc10 : acc11);
      const int n = bn + wnx + sn * 16 + lr;
      const unsigned char ipn = img_pad[b * NDIM + n];
      const float ivy = inv_y[b * NDIM + n];
#pragma unroll
      for (int r = 0; r < 8; ++r) {
        const int m = bm + wmy + sm * 16 + r + 8 * half;
        const unsigned char tpm = txt_pad[b * MDIM + m];
        float dot = acc[r] * inv_x[b * MDIM + m] * ivy;
        float c = 1.0f - dot;
        bool jp = (tpm | ipn) != 0;
        cost[((size_t)b * MDIM + m) * NDIM + n] = jp ? 0.f : c;
        size_t tix = ((size_t)b * NDIM + n) * MDIM + m;
        Amat[tix] = jp ? 0.f : __expf(-2.0f * c);   // beta = 0.5
        Tmat[tix] = jp ? 0.f : 1.0f;
      }
    }
  }
}

// =========================================================================
// 4) IPOT iteration kernels (3 per iteration, 50 iterations)
// =========================================================================
__global__ void ipot_qrow_kernel(const float* __restrict__ Amat,
                                 const float* __restrict__ Tmat,
                                 const float* __restrict__ sigma,
                                 float* __restrict__ Qmat,
                                 float* __restrict__ rvec) {
  const int b = blockIdx.y;
  const int wave = threadIdx.x >> 5, lane = threadIdx.x & 31;
  const float* sg = sigma + b * MDIM;
#pragma unroll
  for (int rr = 0; rr < 2; ++rr) {
    const int n = blockIdx.x * 16 + wave * 2 + rr;
    const size_t base = ((size_t)b * NDIM + n) * MDIM;
    float acc = 0.f;
    for (int m = lane; m < MDIM; m += 32) {
      float q = Amat[base + m] * Tmat[base + m];
      Qmat[base + m] = q;
      acc += q * sg[m];
    }
    for (int off = 16; off > 0; off >>= 1) acc += __shfl_down(acc, off, 32);
    if (lane == 0) rvec[b * NDIM + n] = acc;
  }
}

__global__ void ipot_col_sigma_kernel(const float* __restrict__ Qmat,
                                      const float* __restrict__ rvec,
                                      const unsigned char* __restrict__ txt_pad,
                                      const unsigned char* __restrict__ img_pad,
                                      const float* __restrict__ xlen,
                                      const float* __restrict__ ylen,
                                      float* __restrict__ sigma) {
  __shared__ float ds[NDIM];
  const int b = blockIdx.y, tid = threadIdx.x;
  const float yl = ylen[b];
#pragma unroll
  for (int i = 0; i < 2; ++i) {
    int n = tid + i * 256;
    float ym = img_pad[b * NDIM + n] ? 10000.f : 0.f;
    ds[n] = 1.0f / (yl * rvec[b * NDIM + n] + ym);
  }
  __syncthreads();
  const int m = blockIdx.x * 256 + tid;
  const size_t base = (size_t)b * NDIM * MDIM + m;
  float acc = 0.f;
  for (int n = 0; n < NDIM; ++n) acc += ds[n] * Qmat[base + (size_t)n * MDIM];
  float xm = txt_pad[b * MDIM + m] ? 10000.f : 0.f;
  sigma[b * MDIM + m] = 1.0f / (xlen[b] * acc + xm);
}

__global__ void ipot_update_kernel(const float* __restrict__ Qmat,
                                   const float* __restrict__ rvec,
                                   const float* __restrict__ sigma,
                                   const unsigned char* __restrict__ img_pad,
                                   const float* __restrict__ ylen,
                                   float* __restrict__ Tmat) {
  __shared__ float sg[MDIM];
  const int b = blockIdx.y, tid = threadIdx.x;
  sg[tid] = sigma[b * MDIM + tid];
  sg[tid + 256] = sigma[b * MDIM + tid + 256];
  __syncthreads();
  const int wave = tid >> 5, lane = tid & 31;
  const float yl = ylen[b];
#pragma unroll
  for (int rr = 0; rr < 2; ++rr) {
    const int n = blockIdx.x * 16 + wave * 2 + rr;
    float ym = img_pad[b * NDIM + n] ? 10000.f : 0.f;
    float dn = 1.0f / (yl * rvec[b * NDIM + n] + ym);
    const size_t base = ((size_t)b * NDIM + n) * MDIM;
    for (int m = lane; m < MDIM; m += 32)
      Tmat[base + m] = dn * Qmat[base + m] * sg[m];
  }
}

// =========================================================================
// 5) OT final contraction: sum_{b,m,n} cost[b][m][n] * T[b][n][m]
// =========================================================================
__global__ void zero_acc_kernel(float* __restrict__ ot_acc) { ot_acc[0] = 0.f; }

__global__ void ot_dot_kernel(const float* __restrict__ cost,
                              const float* __restrict__ Tmat,
                              float* __restrict__ ot_acc) {
  __shared__ float Ts[64][65];
  __shared__ float red[8];
  const int b = blockIdx.z;
  const int n0 = blockIdx.x * 64, m0 = blockIdx.y * 64;
  const int tid = threadIdx.x;
#pragma unroll
  for (int i = 0; i < 16; ++i) {
    int idx = tid + i * 256;
    int tn = idx >> 6, tm = idx & 63;
    Ts[tn][tm] = Tmat[((size_t)b * NDIM + n0 + tn) * MDIM + m0 + tm];
  }
  __syncthreads();
  float acc = 0.f;
#pragma unroll
  for (int i = 0; i < 16; ++i) {
    int idx = tid + i * 256;
    int am = idx >> 6, an = idx & 63;
    acc += cost[((size_t)b * MDIM + m0 + am) * NDIM + n0 + an] * Ts[an][am];
  }
  for (int off = 16; off > 0; off >>= 1) acc += __shfl_down(acc, off, 32);
  if ((tid & 31) == 0) red[tid >> 5] = acc;
  __syncthreads();
  if (tid == 0) {
    float s = 0.f;
    for (int w = 0; w < 8; ++w) s += red[w];
    atomicAdd(ot_acc, s);
  }
}

// =========================================================================
// 6) CRF numerator (one thread per batch)
// =========================================================================
__global__ void crf_num_kernel(const float* __restrict__ emis,
                               const int* __restrict__ tags,
                               const unsigned char* __restrict__ mask,
                               const float* __restrict__ startt,
                               const float* __restrict__ endt,
                               const float* __restrict__ trans,
                               float* __restrict__ numer) {
  int b = threadIdx.x;
  int prev = tags[b];
  float score = startt[prev] + emis[(size_t)b * KTAG + prev];
  int last = prev;
  for (int t = 1; t < TSTEP; ++t) {
    int tg = tags[t * BATCH + b];
    if (mask[t * BATCH + b]) {
      score += trans[prev * KTAG + tg] + emis[((size_t)t * BATCH + b) * KTAG + tg];
      last = tg;
    }
    prev = tg;
  }
  numer[b] = score + endt[last];
}

// =========================================================================
// 7) CRF forward (denominator) — one wave per batch; llh = num - den
// =========================================================================
__global__ void crf_den_kernel(const float* __restrict__ emis,
                               const unsigned char* __restrict__ mask,
                               const float* __restrict__ startt,
                               const float* __restrict__ endt,
                               const float* __restrict__ trans,
                               const float* __restrict__ numer,
                               float* __restrict__ llh) {
  __shared__ float tr[KTAG * KTAG];
  __shared__ float ls[32];
  const int b = blockIdx.x, j = threadIdx.x;
  for (int i = j; i < KTAG * KTAG; i += 32) tr[i] = trans[i];
  float s = (j < KTAG) ? startt[j] + emis[(size_t)b * KTAG + j] : -INFINITY;
  __syncthreads();
  for (int t = 1; t < TSTEP; ++t) {
    ls[j] = s;
    __syncthreads();
    if (j < KTAG) {
      float e = emis[((size_t)t * BATCH + b) * KTAG + j];
      float mx = -INFINITY;
      for (int i = 0; i < KTAG; ++i) mx = fmaxf(mx, ls[i] + tr[i * KTAG + j]);
      float sum = 0.f;
      for (int i = 0; i < KTAG; ++i) sum += __expf(ls[i] + tr[i * KTAG + j] - mx);
      float nxt = mx + __logf(sum) + e;
      if (mask[t * BATCH + b]) s = nxt;
    }
    __syncthreads();
  }
  ls[j] = (j < KTAG) ? s + endt[j] : -INFINITY;
  __syncthreads();
  if (j == 0) {
    float mx = -INFINITY;
    for (int i = 0; i < KTAG; ++i) mx = fmaxf(mx, ls[i]);
    float sum = 0.f;
    for (int i = 0; i < KTAG; ++i) sum += __expf(ls[i] - mx);
    llh[b] = numer[b] - (mx + __logf(sum));
  }
}

// =========================================================================
// 8) loss = -sum(llh) + ot_acc
// =========================================================================
__global__ void final_kernel(const float* __restrict__ llh,
                             const float* __restrict__ ot_acc,
                             float* __restrict__ out) {
  __shared__ float sm[64];
  int t = threadIdx.x;
  sm[t] = llh[t];
  __syncthreads();
  for (int off = 32; off > 0; off >>= 1) {
    if (t < off) sm[t] += sm[t + off];
    __syncthreads();
  }
  if (t == 0) out[0] = ot_acc[0] - sm[0];
}

// =========================================================================
extern "C" void kernel_launch(void* const* d_in, const int* in_sizes, int n_in,
                              void* d_out, int out_size, void* d_ws, size_t ws_size,
                              hipStream_t stream) {
  const float* emissions = (const float*)d_in[0];
  const float* txt_emb   = (const float*)d_in[1];
  const float* img_emb   = (const float*)d_in[2];
  const float* startt    = (const float*)d_in[3];
  const float* endt      = (const float*)d_in[4];
  const float* trans     = (const float*)d_in[5];
  const int*   tags      = (const int*)d_in[6];
  const unsigned char* crf_mask = (const unsigned char*)d_in[7];
  const unsigned char* txt_pad  = (const unsigned char*)d_in[8];
  const unsigned char* img_pad  = (const unsigned char*)d_in[9];
  float* out = (float*)d_out;

  float* ws    = (float*)d_ws;
  float* cost  = ws;
  float* Amat  = cost + SZ_MAT;
  float* Tmat  = Amat + SZ_MAT;
  float* Qmat  = Tmat + SZ_MAT;
  float* inv_x = Qmat + SZ_MAT;
  float* inv_y = inv_x + BATCH * MDIM;
  float* sigma = inv_y + BATCH * NDIM;
  float* rvec  = sigma + BATCH * MDIM;
  float* xlen  = rvec + BATCH * NDIM;
  float* ylen  = xlen + BATCH;
  float* numer = ylen + BATCH;
  float* llh   = numer + BATCH;
  float* otacc = llh + BATCH;

  invnorm_kernel<<<dim3((2 * BATCH * MDIM) / 8), dim3(256), 0, stream>>>(
      txt_emb, img_emb, inv_x, inv_y);
  lengths_sigma_kernel<<<dim3(BATCH), dim3(512), 0, stream>>>(
      txt_pad, img_pad, xlen, ylen, sigma);

  cost_wmma_kernel<<<dim3(MDIM / 64, NDIM / 64, BATCH), dim3(128), 0, stream>>>(
      txt_emb, img_emb, inv_x, inv_y, txt_pad, img_pad, cost, Amat, Tmat);

  zero_acc_kernel<<<dim3(1), dim3(1), 0, stream>>>(otacc);

  for (int it = 0; it < OT_ITERS; ++it) {
    ipot_qrow_kernel<<<dim3(NDIM / 16, BATCH), dim3(256), 0, stream>>>(
        Amat, Tmat, sigma, Qmat, rvec);
    ipot_col_sigma_kernel<<<dim3(MDIM / 256, BATCH), dim3(256), 0, stream>>>(
        Qmat, rvec, txt_pad, img_pad, xlen, ylen, sigma);
    ipot_update_kernel<<<dim3(NDIM / 16, BATCH), dim3(256), 0, stream>>>(
        Qmat, rvec, sigma, img_pad, ylen, Tmat);
  }

  ot_dot_kernel<<<dim3(NDIM / 64, MDIM / 64, BATCH), dim3(256), 0, stream>>>(
      cost, Tmat, otacc);

  crf_num_kernel<<<dim3(1), dim3(BATCH), 0, stream>>>(
      emissions, tags, crf_mask, startt, endt, trans, numer);
  crf_den_kernel<<<dim3(BATCH), dim3(32), 0, stream>>>(
      emissions, crf_mask, startt, endt, trans, numer, llh);

  final_kernel<<<dim3(1), dim3(64), 0, stream>>>(llh, otacc, out);
}